// RotaryPositionalEmbedding_56916906607007
// MI455X (gfx1250) — compile-verified
//
#include <hip/hip_runtime.h>
#include <hip/hip_bf16.h>
#include <stdint.h>

#define AS_GLOBAL __attribute__((address_space(1)))
#define AS_LOCAL  __attribute__((address_space(3)))

typedef float v4f __attribute__((ext_vector_type(4)));

// RoPE: out[b,h,s,:] = M[pos[b,s]] @ x[b,h,s,:], M block-diagonal 2x2 rotations.
// B=4, H=16, S=2048, D=64. Memory-bound: ~67MB streaming traffic -> ~2.9us at
// 23.3TB/s. Streaming x/out use TH=NT so the 33.5MB rope-matrix table stays
// L2-resident for the gathers; gathers go straight to LDS via the gfx1250
// async-to-LDS path and are reused by all 16 heads of a (b,s).
__global__ __launch_bounds__(256) void rope_rotate_kernel(
    const float* __restrict__ x,          // (B,H,S,D) f32
    const int*   __restrict__ tok_pos,    // (B,S) i32
    const float* __restrict__ rope_m,     // (S_max, D, D) f32
    float*       __restrict__ out)        // (B,H,S,D) f32
{
    constexpr int H = 16, S = 2048, D = 64;

    // cs[0..31]  = cos_k  (M[p, 2k,   2k])
    // cs[32..63] = sin_k  (M[p, 2k+1, 2k])
    __shared__ float cs[64];

    const int bs  = blockIdx.x;        // b*S + s
    const int b   = bs >> 11;
    const int s   = bs & (S - 1);
    const int tid = threadIdx.x;

    // Block-uniform position lookup (scalar load), clip a_max = S-1.
    int p = tok_pos[bs];
    p = (p > (S - 1)) ? (S - 1) : p;

    // Waves 0 and 1: per-lane async gather of cos/sin directly into LDS.
    if (tid < 64) {
        const int k     = tid & 31;     // rotation-pair index
        const int isSin = tid >> 5;     // wave0 -> cos row 2k, wave1 -> sin row 2k+1
        const size_t gidx = (size_t)p * (D * D)
                          + (size_t)(2 * k + isSin) * D
                          + (size_t)(2 * k);
#if __has_builtin(__builtin_amdgcn_global_load_async_to_lds_b32)
        __builtin_amdgcn_global_load_async_to_lds_b32(
            (AS_GLOBAL int*)(rope_m + gidx),
            (AS_LOCAL int*)&cs[tid],
            0 /*imm offset*/, 0 /*cpol: RT, keep table cached*/);
#else
        {
            uint64_t gaddr = (uint64_t)(uintptr_t)(rope_m + gidx);
            uint32_t laddr = (uint32_t)(uintptr_t)(AS_LOCAL float*)&cs[tid];
            asm volatile("global_load_async_to_lds_b32 %0, %1, off"
                         :: "v"(laddr), "v"(gaddr)
                         : "memory");
        }
#endif
    }

    // Drain ASYNCcnt (no-op for waves 2..7), then make LDS visible block-wide.
#if __has_builtin(__builtin_amdgcn_s_wait_asynccnt)
    __builtin_amdgcn_s_wait_asynccnt(0);
#else
    asm volatile("s_wait_asynccnt 0" ::: "memory");
#endif
    __syncthreads();

    // 16 threads per head-row; each thread rotates 2 pairs via one b128 ld/st.
    const int h  = tid >> 4;
    const int ln = tid & 15;
    const size_t rowq = ((size_t)(b * H + h) * S + s) * (D / 4) + (size_t)ln;

    // Read-once stream: non-temporal load (TH_LOAD_NT).
    const v4f v = __builtin_nontemporal_load((const v4f*)x + rowq);

    const int k0 = 2 * ln;
    const int k1 = 2 * ln + 1;
    const float c0 = cs[k0],  s0 = cs[32 + k0];
    const float c1 = cs[k1],  s1 = cs[32 + k1];

    v4f r;
    r.x = c0 * v.x - s0 * v.y;
    r.y = s0 * v.x + c0 * v.y;
    r.z = c1 * v.z - s1 * v.w;
    r.w = s1 * v.z + c1 * v.w;

    // Write-once stream: non-temporal store (TH_STORE_NT).
    __builtin_nontemporal_store(r, (v4f*)out + rowq);
}

extern "C" void kernel_launch(void* const* d_in, const int* in_sizes, int n_in,
                              void* d_out, int out_size, void* d_ws, size_t ws_size,
                              hipStream_t stream) {
    const float* x   = (const float*)d_in[0];   // (4,16,2048,64) f32
    const int*   pos = (const int*)d_in[1];     // (4,2048) i32
    const float* M   = (const float*)d_in[2];   // (2048,64,64) f32
    float*       out = (float*)d_out;

    const int num_bs = in_sizes[1];             // B*S = 8192
    rope_rotate_kernel<<<dim3(num_bs), dim3(256), 0, stream>>>(x, pos, M, out);
    (void)n_in; (void)out_size; (void)d_ws; (void)ws_size;
}